// EnhancedHeterogeneousLoss_35235911696998
// MI455X (gfx1250) — compile-verified
//
#include <hip/hip_runtime.h>
#include <math.h>

// ---------------- problem constants (structure of the reference) ------------
#define BGRAPH 4096
#define FDIM   66
#define NDEV   34
#define EPSF   1e-8f

typedef float v2f __attribute__((ext_vector_type(2)));
typedef float v8f __attribute__((ext_vector_type(8)));

// ---------------- workspace layout (in floats) ------------------------------
#define OFF_FEAT 0                         // 4096 x 66 segment feature sums
#define OFF_CNT  (BGRAPH * FDIM)           // 4096 counts (as f32)
#define OFF_PG   (OFF_CNT + BGRAPH)        // 9 x 4096 per-graph quantities
#define NQ       9

// ---------------- kernel 0: zero the accumulators ---------------------------
__global__ void ehl_zero(float* __restrict__ ws, int n) {
  int i = blockIdx.x * blockDim.x + threadIdx.x;
  if (i < n) ws[i] = 0.f;
}

// ---------------- kernel 1: streaming segment-sum (bandwidth bound) ---------
// One wave owns RPW contiguous rows. batch is sorted, so the wave accumulates
// feature partial sums in registers and only flushes on a segment change.
// Lane L accumulates columns 2L, 2L+1 (float2); lane 0 also columns 64,65.
#define RPW 128

__device__ __forceinline__ void ehl_flush(float* __restrict__ feat,
                                          float* __restrict__ cnts,
                                          int seg, int lane,
                                          v2f acc, v2f acc2, int cnt) {
  float* row = feat + (long)seg * FDIM;
  atomicAdd(row + 2 * lane,     acc.x);
  atomicAdd(row + 2 * lane + 1, acc.y);
  if (lane == 0) {
    atomicAdd(row + 64, acc2.x);
    atomicAdd(row + 65, acc2.y);
    atomicAdd(cnts + seg, (float)cnt);
  }
}

__global__ void ehl_segsum(const float* __restrict__ x,
                           const int*   __restrict__ batch,
                           float*       __restrict__ ws, int N) {
  const int lane = threadIdx.x & 31;
  const int gw   = blockIdx.x * (blockDim.x >> 5) + (threadIdx.x >> 5);
  long r0 = (long)gw * RPW;
  if (r0 >= N) return;
  long r1 = r0 + RPW; if (r1 > N) r1 = N;

  float* feat = ws + OFF_FEAT;
  float* cnts = ws + OFF_CNT;

  const v2f vzero = {0.f, 0.f};
  v2f acc  = vzero;
  v2f acc2 = vzero;          // cols 64,65 (lane 0 only)
  int cnt = 0;
  int curSeg = -1;

  for (long r = r0; r < r1; ++r) {
    int seg = batch[r];                       // wave-uniform -> scalar load
    if (seg != curSeg) {
      if (cnt > 0) {
        ehl_flush(feat, cnts, curSeg, lane, acc, acc2, cnt);
        acc = vzero; acc2 = vzero; cnt = 0;
      }
      curSeg = seg;
    }
    const v2f* xr = (const v2f*)(x + r * (long)FDIM);
    v2f v = xr[lane];                         // coalesced 256B per wave
    acc += v;
    if (lane == 0) acc2 += xr[32];            // tail columns 64,65
    cnt++;
    long pf = r + 8;                          // software prefetch (CDNA5 path)
    if (pf < r1) __builtin_prefetch(x + pf * (long)FDIM + 2 * lane, 0, 0);
  }
  if (cnt > 0) ehl_flush(feat, cnts, curSeg, lane, acc, acc2, cnt);
}

// ---------------- kernel 2: per-graph scalar math (tiny) --------------------
__global__ void ehl_graph(const float* __restrict__ pred,
                          const int*   __restrict__ tgt,
                          const float* __restrict__ dw,
                          float*       __restrict__ ws) {
  int b = blockIdx.x * blockDim.x + threadIdx.x;
  if (b >= BGRAPH) return;

  const float* feat = ws + OFF_FEAT;
  const float* cnts = ws + OFF_CNT;
  float*       pg   = ws + OFF_PG;

  float l  = pred[b];
  float tf = (float)tgt[b];
  float bce   = fmaxf(l, 0.f) - l * tf + log1pf(expf(-fabsf(l)));
  float probs = 1.f / (1.f + expf(-l));
  float unc   = 1.f - fabsf(probs - 0.5f) * 2.f;
  float cnt   = cnts[b];

  const float* fs = feat + (long)b * FDIM;

  // device distribution: S, S*w, entropy over 34 devices
  float sarr[NDEV];
  float S = 0.f, Sw = 0.f;
#pragma unroll
  for (int i = 0; i < NDEV; ++i) {
    float s = fs[i] + EPSF;
    sarr[i] = s;
    S  += s;
    Sw += s * dw[i];
  }
  float ent = 0.f;
#pragma unroll
  for (int i = 0; i < NDEV; ++i) {
    float d = sarr[i] / S;
    ent -= d * logf(d + EPSF);
  }
  float imp = Sw / S;
  float err = fabsf(probs - tf);
  float pgd = err * imp + ent * err * 0.15f;

  // cosine similarity of means[:, :32] vs means[:, 34:66]
  float safe = fmaxf(cnt, 1.f);
  float dot = 0.f, na = 0.f, nb = 0.f;
#pragma unroll
  for (int k = 0; k < 32; ++k) {
    float a = fs[k]        / safe;
    float h = fs[34 + k]   / safe;
    dot = fmaf(a, h, dot);
    na  = fmaf(a, a, na);
    nb  = fmaf(h, h, nb);
  }
  float fsim = fabsf(dot / (fmaxf(sqrtf(na), EPSF) * fmaxf(sqrtf(nb), EPSF)));
  float conf = fabsf(probs - 0.5f) * 2.f;
  float cons = fabsf(fsim - conf) * 1.2f;

  float pos = (tf == 1.f) ? 1.f : 0.f;
  float neg = (tf == 0.f) ? 1.f : 0.f;

  pg[0 * BGRAPH + b] = bce * (1.f + 2.f * unc);
  pg[1 * BGRAPH + b] = (cnt > 0.f)  ? pgd  : 0.f;
  pg[2 * BGRAPH + b] = (cnt > 0.f)  ? 1.f  : 0.f;
  pg[3 * BGRAPH + b] = (cnt >= 2.f) ? cons : 0.f;
  pg[4 * BGRAPH + b] = (cnt >= 2.f) ? 1.f  : 0.f;
  pg[5 * BGRAPH + b] = fmaxf(0.6f - probs, 0.f) * pos;
  pg[6 * BGRAPH + b] = pos;
  pg[7 * BGRAPH + b] = fmaxf(probs - 0.4f, 0.f) * neg;
  pg[8 * BGRAPH + b] = neg;
}

// ---------------- kernel 3: WMMA reduction 9 x (4096 -> scalar) + combine ---
// V_WMMA_F32_16X16X4_F32 with B = ones: D[m][n] += sum_k A[m][k] (exact f32).
// A-layout: lane L<16 -> (A[L][0],A[L][1]); lane L>=16 -> (A[L-16][2],A[L-16][3]).
// C-layout: VGPR j, lanes 0-15 = D[j][n], lanes 16-31 = D[8+j][n].
__global__ void ehl_reduce(const float* __restrict__ ws, float* __restrict__ out) {
  __shared__ float sred[NQ * 2];
  const int lane = threadIdx.x & 31;
  const int q    = threadIdx.x >> 5;            // wave id = quantity id, 0..8

  const float* pg = ws + OFF_PG + (long)q * BGRAPH;

  v8f c = {};                                    // f32 accumulator tile
  v2f bones = {1.f, 1.f};                        // B all-ones (4x16)
  const int idx = (lane & 15) * 4 + ((lane >> 4) ? 2 : 0);

#pragma unroll 4
  for (int t = 0; t < BGRAPH / 64; ++t) {        // 64 values per WMMA
    v2f a = *(const v2f*)(pg + t * 64 + idx);
    c = __builtin_amdgcn_wmma_f32_16x16x4_f32(
            /*neg_a=*/false, a, /*neg_b=*/false, bones,
            /*c_mod=*/(short)0, c, /*reuse_a=*/false, /*reuse_b=*/false);
  }
  // column 0 lives in lane 0 (rows 0-7) and lane 16 (rows 8-15)
  float p = c[0] + c[1] + c[2] + c[3] + c[4] + c[5] + c[6] + c[7];
  if ((lane & 15) == 0) sred[q * 2 + (lane >> 4)] = p;
  __syncthreads();

  if (threadIdx.x == 0) {
    float S[NQ];
#pragma unroll
    for (int i = 0; i < NQ; ++i) S[i] = sred[2 * i] + sred[2 * i + 1];
    float base = S[0] * (1.f / (float)BGRAPH);
    float dev  = S[1] / fmaxf(S[2], 1.f);
    float topo = S[3] / fmaxf(S[4], 1.f);
    float posl = S[5] / fmaxf(S[6], 1.f);
    float negl = S[7] / fmaxf(S[8], 1.f);
    float contrast = (S[6] > 0.f && S[8] > 0.f) ? (posl + negl) : 0.f;
    out[0] = 0.4f * base + 0.35f * dev + 0.2f * topo + 0.05f * contrast;
  }
}

// ---------------- host launcher ---------------------------------------------
extern "C" void kernel_launch(void* const* d_in, const int* in_sizes, int n_in,
                              void* d_out, int out_size, void* d_ws, size_t ws_size,
                              hipStream_t stream) {
  const float* pred  = (const float*)d_in[0];
  const int*   tgt   = (const int*)  d_in[1];
  const float* x     = (const float*)d_in[2];
  const int*   batch = (const int*)  d_in[3];
  const float* dw    = (const float*)d_in[4];
  float* ws  = (float*)d_ws;
  float* out = (float*)d_out;
  const int N = in_sizes[3];                     // 1,000,000 rows

  // 0) zero feat-sum + count accumulators
  const int nz = OFF_PG;
  ehl_zero<<<(nz + 255) / 256, 256, 0, stream>>>(ws, nz);

  // 1) streaming segment sum: wave-per-RPW-rows, 8 waves per 256-thread block
  const int waves  = (N + RPW - 1) / RPW;
  const int blocks = (waves + 7) / 8;
  ehl_segsum<<<blocks, 256, 0, stream>>>(x, batch, ws, N);

  // 2) per-graph math
  ehl_graph<<<BGRAPH / 256, 256, 0, stream>>>(pred, tgt, dw, ws);

  // 3) WMMA reductions + final combine (9 waves, one block)
  ehl_reduce<<<1, NQ * 32, 0, stream>>>(ws, out);
}